// MS_MSA_14096082665914
// MI455X (gfx1250) — compile-verified
//
#include <hip/hip_runtime.h>
#include <hip/hip_bf16.h>
#include <math.h>

typedef __attribute__((ext_vector_type(2))) float v2f;
typedef __attribute__((ext_vector_type(8))) float v8f;

#define B_     2
#define HH_    256
#define WW_    256
#define C_     256
#define HEADS_ 8
#define DH_    32
#define NPIX_  (HH_ * WW_)               // 65536
#define NB_    ((size_t)B_ * NPIX_ * C_) // 33554432 floats per big buffer

__device__ __forceinline__ float gelu_exact(float x) {
  return 0.5f * x * (1.0f + erff(x * 0.70710678118654752f));
}

// ---------------------------------------------------------------------------
// small zero-fill for accumulator buffers (sums + attn_raw)
// ---------------------------------------------------------------------------
__global__ void k_zero(float* __restrict__ p, int n) {
  int i = blockIdx.x * blockDim.x + threadIdx.x;
  if (i < n) p[i] = 0.0f;
}

// ---------------------------------------------------------------------------
// depthwise 3x3 conv, NHWC, per-channel weights (256x9).
// mode 0: out = gelu(conv + bias)
// mode 1: out = (conv + bias) + resid      (positional-embedding residual)
// one block = one pixel, 256 threads = channels (coalesced NHWC access)
// ---------------------------------------------------------------------------
__global__ void k_dwconv_pos(const float* __restrict__ in,
                             const float* __restrict__ wgt,
                             const float* __restrict__ bias,
                             const float* __restrict__ resid,
                             float* __restrict__ out, int mode) {
  int c   = threadIdx.x;
  int pix = blockIdx.x;                 // b*65536 + y*256 + x
  int b   = pix >> 16;
  int y   = (pix >> 8) & 255;
  int x   = pix & 255;
  float w[9];
#pragma unroll
  for (int j = 0; j < 9; ++j) w[j] = wgt[c * 9 + j];
  float acc = bias[c];
#pragma unroll
  for (int dy = 0; dy < 3; ++dy) {
    int yy = y + dy - 1;
    if ((unsigned)yy >= 256u) continue;
#pragma unroll
    for (int dx = 0; dx < 3; ++dx) {
      int xx = x + dx - 1;
      if ((unsigned)xx >= 256u) continue;
      acc += in[((size_t)b * NPIX_ + yy * 256 + xx) * C_ + c] * w[dy * 3 + dx];
    }
  }
  if (mode == 0) acc = gelu_exact(acc);
  else           acc += resid[(size_t)pix * C_ + c];
  out[(size_t)pix * C_ + c] = acc;
}

// ---------------------------------------------------------------------------
// fused chain of three depthwise 3x3 convs (per-head shared weights).
// Tile: 16x16 output pixels x 8 channels (all same head). Halo tiling:
// input 22x22 -> conv1 20x20 -> conv2 18x18 -> conv3 16x16 (all in LDS).
// Intermediate pixels outside the image are forced to 0 to match SAME-pad
// semantics of sequential convs. Output written in (b, c, n) layout for the
// downstream WMMA GEMMs. Also reduces sum(x) or sum(x^2) per (b,c) channel
// via wave32 shuffles + one atomic per wave (feeds L2-norms / v-mean).
// ---------------------------------------------------------------------------
__global__ void k_chain3(const float* __restrict__ x1,
                         const float* __restrict__ w1g,
                         const float* __restrict__ w2g,
                         const float* __restrict__ w3g,
                         float* __restrict__ out,
                         float* __restrict__ sums, int squared) {
  __shared__ float s_in[22 * 22 * 8];
  __shared__ float s_b1[20 * 20 * 8];
  __shared__ float s_b2[18 * 18 * 8];
  int tid  = threadIdx.x;
  int cg   = blockIdx.z & 31;     // channel group of 8
  int b    = blockIdx.z >> 5;
  int head = cg >> 2;             // (cg*8)/32
  int x0 = blockIdx.x * 16, y0 = blockIdx.y * 16;
  float w1[9], w2[9], w3[9];
#pragma unroll
  for (int j = 0; j < 9; ++j) {
    w1[j] = w1g[head * 9 + j];
    w2[j] = w2g[head * 9 + j];
    w3[j] = w3g[head * 9 + j];
  }
  // load input halo tile
  for (int i = tid; i < 22 * 22 * 8; i += 256) {
    int c = i & 7; int p = i >> 3; int px = p % 22; int py = p / 22;
    int gy = y0 + py - 3, gx = x0 + px - 3;
    float v = 0.0f;
    if ((unsigned)gy < 256u && (unsigned)gx < 256u)
      v = x1[((size_t)b * NPIX_ + gy * 256 + gx) * C_ + cg * 8 + c];
    s_in[i] = v;
  }
  __syncthreads();
  // conv1 -> 20x20
  for (int i = tid; i < 20 * 20 * 8; i += 256) {
    int c = i & 7; int p = i >> 3; int px = p % 20; int py = p / 20;
    int gy = y0 + py - 2, gx = x0 + px - 2;
    float a = 0.0f;
    if ((unsigned)gy < 256u && (unsigned)gx < 256u) {
#pragma unroll
      for (int dy = 0; dy < 3; ++dy)
#pragma unroll
        for (int dx = 0; dx < 3; ++dx)
          a += s_in[((py + dy) * 22 + (px + dx)) * 8 + c] * w1[dy * 3 + dx];
    }
    s_b1[i] = a;
  }
  __syncthreads();
  // conv2 -> 18x18
  for (int i = tid; i < 18 * 18 * 8; i += 256) {
    int c = i & 7; int p = i >> 3; int px = p % 18; int py = p / 18;
    int gy = y0 + py - 1, gx = x0 + px - 1;
    float a = 0.0f;
    if ((unsigned)gy < 256u && (unsigned)gx < 256u) {
#pragma unroll
      for (int dy = 0; dy < 3; ++dy)
#pragma unroll
        for (int dx = 0; dx < 3; ++dx)
          a += s_b1[((py + dy) * 20 + (px + dx)) * 8 + c] * w2[dy * 3 + dx];
    }
    s_b2[i] = a;
  }
  __syncthreads();
  // conv3 -> 16x16 + transposed (b,c,n) store + channel reduction
  int c  = tid >> 5;      // wave-uniform channel
  int ln = tid & 31;
  float local = 0.0f;
  for (int it = 0; it < 8; ++it) {
    int p = ln + it * 32; int py = p >> 4; int px = p & 15;
    float a = 0.0f;
#pragma unroll
    for (int dy = 0; dy < 3; ++dy)
#pragma unroll
      for (int dx = 0; dx < 3; ++dx)
        a += s_b2[((py + dy) * 18 + (px + dx)) * 8 + c] * w3[dy * 3 + dx];
    int n = (y0 + py) * 256 + (x0 + px);
    out[((size_t)b * C_ + cg * 8 + c) * NPIX_ + n] = a;
    local += squared ? a * a : a;
  }
  for (int off = 16; off > 0; off >>= 1) local += __shfl_down(local, off, 32);
  if (ln == 0) atomicAdd(&sums[b * C_ + cg * 8 + c], local);
}

// ---------------------------------------------------------------------------
// attn_raw[b,h,d,e] = sum_n k[d,n]*q[e,n]  via V_WMMA_F32_16X16X4_F32.
// One wave per block; each wave owns a 256-wide n-chunk of one (b,h) and
// accumulates a 32x32 tile (2x2 WMMA tiles), then f32-atomically merges.
// Normalization by ||k_d||*||q_e|| factors out and is applied in finalize.
// ---------------------------------------------------------------------------
__global__ void k_attn_wmma(const float* __restrict__ q,
                            const float* __restrict__ k,
                            float* __restrict__ attn_raw) {
  const int CH = 256;
  int blk   = blockIdx.x;
  int chunk = blk & 255;
  int bh    = blk >> 8;           // b*8 + h
  int lane  = threadIdx.x & 31;
  int half  = lane >> 4, lm = lane & 15;
  int n0    = chunk * CH;
  size_t rowbase = (size_t)bh * 32;   // == b*256 + h*32
  const float* kr0 = k + (rowbase + lm) * (size_t)NPIX_;
  const float* kr1 = k + (rowbase + 16 + lm) * (size_t)NPIX_;
  const float* qr0 = q + (rowbase + lm) * (size_t)NPIX_;
  const float* qr1 = q + (rowbase + 16 + lm) * (size_t)NPIX_;
  int off = 2 * half;   // lanes 0-15 hold K=0,1 ; lanes 16-31 hold K=2,3
  v8f c00 = {}, c01 = {}, c10 = {}, c11 = {};
  for (int n = n0; n < n0 + CH; n += 4) {
    v2f a0 = *reinterpret_cast<const v2f*>(kr0 + n + off);
    v2f a1 = *reinterpret_cast<const v2f*>(kr1 + n + off);
    v2f b0 = *reinterpret_cast<const v2f*>(qr0 + n + off);
    v2f b1 = *reinterpret_cast<const v2f*>(qr1 + n + off);
    c00 = __builtin_amdgcn_wmma_f32_16x16x4_f32(false, a0, false, b0, (short)0, c00, false, false);
    c01 = __builtin_amdgcn_wmma_f32_16x16x4_f32(false, a0, false, b1, (short)0, c01, false, false);
    c10 = __builtin_amdgcn_wmma_f32_16x16x4_f32(false, a1, false, b0, (short)0, c10, false, false);
    c11 = __builtin_amdgcn_wmma_f32_16x16x4_f32(false, a1, false, b1, (short)0, c11, false, false);
  }
  float* ab = attn_raw + (size_t)bh * 32 * 32;
#pragma unroll
  for (int r = 0; r < 8; ++r) {
    int d = r + 8 * half;           // C/D layout: VGPR r -> M=r (+8 upper half)
    atomicAdd(ab + d * 32 + lm,             c00[r]);
    atomicAdd(ab + d * 32 + lm + 16,        c01[r]);
    atomicAdd(ab + (d + 16) * 32 + lm,      c10[r]);
    atomicAdd(ab + (d + 16) * 32 + lm + 16, c11[r]);
  }
}

// ---------------------------------------------------------------------------
// finalize: v-mean -> conv1d(heads channels, k=3, SAME) diag term, cosine
// normalization, rescale, row softmax. One wave per (b,h); thread = row d.
// ---------------------------------------------------------------------------
__global__ void k_finalize(const float* __restrict__ attn_raw,
                           const float* __restrict__ sums,
                           const float* __restrict__ c1w,
                           const float* __restrict__ c1b,
                           const float* __restrict__ rescale,
                           const float* __restrict__ rescale2,
                           float* __restrict__ attn_soft) {
  int bh = blockIdx.x; int b = bh >> 3; int h = bh & 7;
  int d  = threadIdx.x;               // 0..31
  const float* sq_q = sums;
  const float* sq_k = sums + 512;
  const float* sv   = sums + 1024;
  float av = c1b[h];
  for (int hi = 0; hi < 8; ++hi)
#pragma unroll
    for (int kk = 0; kk < 3; ++kk) {
      int p = d + kk - 1;
      if ((unsigned)p < 32u)
        av += c1w[(h * 8 + hi) * 3 + kk] * sv[b * 256 + hi * 32 + p] * (1.0f / 65536.0f);
    }
  float nk  = fmaxf(sqrtf(sq_k[b * 256 + h * 32 + d]), 1e-12f);
  float rs  = rescale[h], rs2 = rescale2[h];
  const float* ar = attn_raw + (size_t)bh * 1024 + d * 32;
  float row[32];
  float mx = -3.4e38f;
  for (int e = 0; e < 32; ++e) {
    float nq = fmaxf(sqrtf(sq_q[b * 256 + h * 32 + e]), 1e-12f);
    float a  = ar[e] / (nk * nq) * rs;
    if (e == d) a += av * rs2;
    row[e] = a; mx = fmaxf(mx, a);
  }
  float s = 0.0f;
  for (int e = 0; e < 32; ++e) { row[e] = expf(row[e] - mx); s += row[e]; }
  float inv = 1.0f / s;
  float* o = attn_soft + (size_t)bh * 1024 + d * 32;
  for (int e = 0; e < 32; ++e) o[e] = row[e] * inv;
}

// ---------------------------------------------------------------------------
// W2[b,co,h*32+e] = sum_d proj_w[co, h*32+d] * attn[b,h,d,e]
// folds the 256x256 projection with the block-diagonal attn matmul so the
// final stage is a single big GEMM. One block per (b,co), 256 threads = c'.
// ---------------------------------------------------------------------------
__global__ void k_w2(const float* __restrict__ attn_soft,
                     const float* __restrict__ pw,
                     float* __restrict__ W2) {
  int b  = blockIdx.x >> 8; int co = blockIdx.x & 255;
  int cp = threadIdx.x; int h = cp >> 5; int e = cp & 31;
  const float* as  = attn_soft + (size_t)(b * 8 + h) * 1024;
  const float* pwr = pw + co * 256 + h * 32;
  float s = 0.0f;
#pragma unroll
  for (int d = 0; d < 32; ++d) s += pwr[d] * as[d * 32 + e];
  W2[((size_t)b * 256 + co) * 256 + cp] = s;
}

// ---------------------------------------------------------------------------
// final GEMM: out[b,n,co] = sum_c v[b,c,n] * W2[b,co,c] + proj_b[co]
// via V_WMMA_F32_16X16X4_F32. One wave per 16n x 16co tile, K looped 256/4.
// ---------------------------------------------------------------------------
__global__ void k_gemm(const float* __restrict__ vv,
                       const float* __restrict__ W2,
                       const float* __restrict__ pb,
                       float* __restrict__ out) {
  int wid  = blockIdx.x * (blockDim.x >> 5) + (threadIdx.x >> 5);
  int lane = threadIdx.x & 31;
  int half = lane >> 4, lm = lane & 15;
  int co0 = (wid & 15) << 4;
  int t   = wid >> 4;
  int n0  = (t & 4095) << 4;
  int b   = t >> 12;
  const float* vbase = vv + (size_t)b * C_ * NPIX_;
  const float* wbase = W2 + (size_t)b * C_ * C_;
  v8f acc = {};
  for (int k0 = 0; k0 < 256; k0 += 4) {
    int ka = k0 + 2 * half;
    v2f a, bm;
    a.x = vbase[(size_t)ka * NPIX_ + n0 + lm];          // A[m=lm][K=2h+0]
    a.y = vbase[(size_t)(ka + 1) * NPIX_ + n0 + lm];    // A[m=lm][K=2h+1]
    bm  = *reinterpret_cast<const v2f*>(&wbase[(co0 + lm) * 256 + ka]); // B[K][n=lm]
    acc = __builtin_amdgcn_wmma_f32_16x16x4_f32(false, a, false, bm, (short)0, acc, false, false);
  }
#pragma unroll
  for (int r = 0; r < 8; ++r) {
    int n  = n0 + r + 8 * half;
    int co = co0 + lm;
    out[((size_t)b * NPIX_ + n) * C_ + co] = acc[r] + pb[co];
  }
}

// ---------------------------------------------------------------------------
extern "C" void kernel_launch(void* const* d_in, const int* in_sizes, int n_in,
                              void* d_out, int out_size, void* d_ws, size_t ws_size,
                              hipStream_t stream) {
  (void)in_sizes; (void)n_in; (void)out_size; (void)ws_size;
  const float* x_in  = (const float*)d_in[0];
  const float* qw1   = (const float*)d_in[1];
  const float* qw2   = (const float*)d_in[2];
  const float* qw3   = (const float*)d_in[3];
  const float* kw1   = (const float*)d_in[4];
  const float* kw2   = (const float*)d_in[5];
  const float* kw3   = (const float*)d_in[6];
  const float* vw1   = (const float*)d_in[7];
  const float* vw2   = (const float*)d_in[8];
  const float* vw3   = (const float*)d_in[9];
  const float* posw1 = (const float*)d_in[10];
  const float* posb1 = (const float*)d_in[11];
  const float* posw2 = (const float*)d_in[12];
  const float* posb2 = (const float*)d_in[13];
  const float* c1w   = (const float*)d_in[14];
  const float* c1b   = (const float*)d_in[15];
  const float* rsc   = (const float*)d_in[16];
  const float* rsc2  = (const float*)d_in[17];
  const float* pw    = (const float*)d_in[18];
  const float* pb    = (const float*)d_in[19];
  float* out = (float*)d_out;
  float* ws  = (float*)d_ws;

  float* bufA      = ws;                 // t (pos-emb), then reused as q
  float* x1        = ws + NB_;
  float* bufK      = ws + 2 * NB_;
  float* bufV      = ws + 3 * NB_;
  float* sums      = ws + 4 * NB_;       // [0:512) q^2, [512:1024) k^2, [1024:1536) v
  float* attn_raw  = sums + 1536;        // 2*8*32*32
  float* attn_soft = attn_raw + 16384;
  float* W2        = attn_soft + 16384;

  k_zero<<<(1536 + 16384 + 255) / 256, 256, 0, stream>>>(sums, 1536 + 16384);
  k_dwconv_pos<<<B_ * NPIX_, 256, 0, stream>>>(x_in, posw1, posb1, nullptr, bufA, 0);
  k_dwconv_pos<<<B_ * NPIX_, 256, 0, stream>>>(bufA, posw2, posb2, x_in, x1, 1);
  dim3 cgrid(16, 16, B_ * 32);
  k_chain3<<<cgrid, 256, 0, stream>>>(x1, qw1, qw2, qw3, bufA, sums,        1);
  k_chain3<<<cgrid, 256, 0, stream>>>(x1, kw1, kw2, kw3, bufK, sums + 512,  1);
  k_chain3<<<cgrid, 256, 0, stream>>>(x1, vw1, vw2, vw3, bufV, sums + 1024, 0);
  k_attn_wmma<<<B_ * HEADS_ * 256, 32, 0, stream>>>(bufA, bufK, attn_raw);
  k_finalize<<<B_ * HEADS_, 32, 0, stream>>>(attn_raw, sums, c1w, c1b, rsc, rsc2, attn_soft);
  k_w2<<<B_ * 256, 256, 0, stream>>>(attn_soft, pw, W2);
  k_gemm<<<(B_ * (NPIX_ / 16) * (C_ / 16)) / 8, 256, 0, stream>>>(bufV, W2, pb, out);
}